// GATLayer_25142738550918
// MI455X (gfx1250) — compile-verified
//
#include <hip/hip_runtime.h>
#include <math.h>

#define N_NODES 50000
#define F_DIM   128
#define HEADS   4
#define D_HID   32
#define NEG_SLOPE 0.2f
#define ORD_NEG_INF 0x007FFFFFu   // ord(-inf) = ~0xFF800000

typedef __attribute__((ext_vector_type(2))) float v2f;
typedef __attribute__((ext_vector_type(8))) float v8f;

__device__ __forceinline__ unsigned ordf(float f) {
  unsigned u = __float_as_uint(f);
  return (u & 0x80000000u) ? ~u : (u | 0x80000000u);
}
__device__ __forceinline__ float unordf(unsigned o) {
  unsigned u = (o & 0x80000000u) ? (o & 0x7FFFFFFFu) : ~o;
  return __uint_as_float(u);
}
__device__ __forceinline__ float lrelu(float x) {
  return x > 0.f ? x : NEG_SLOPE * x;
}

// ---------------------------------------------------------------------------
// z[N,128] = x[N,128] @ w[128,128]  via V_WMMA_F32_16X16X4_F32 (exact fp32).
// Block = 256 threads = 8 waves; block owns 16 rows; wave w -> cols [16w,16w+16).
// X tile staged to LDS with GLOBAL_LOAD_ASYNC_TO_LDS_B128 (ASYNCcnt path).
// ---------------------------------------------------------------------------
__global__ __launch_bounds__(256) void gemm128_wmma_f32(
    const float* __restrict__ x, const float* __restrict__ w,
    float* __restrict__ z) {
  __shared__ float xs[16 * F_DIM];           // 8 KB X tile
  const int block_m = blockIdx.x * 16;
  const int tid  = threadIdx.x;
  const int wave = tid >> 5;                 // 0..7 == tile_n
  const int lane = tid & 31;

  // Async DMA of 16x128 fp32 tile (512 x 16B chunks; 2 per thread).
  {
    const float* gsrc = x + (size_t)block_m * F_DIM;
    unsigned lbase = (unsigned)(size_t)(&xs[0]);   // LDS byte offset (low 32 bits)
    #pragma unroll
    for (int i = 0; i < 2; ++i) {
      const int idx4 = tid + i * 256;              // 16B chunk index
      const unsigned laddr = lbase + idx4 * 16;
      const float* g = gsrc + idx4 * 4;
      asm volatile("global_load_async_to_lds_b128 %0, %1, off"
                   :: "v"(laddr), "v"(g)
                   : "memory");
    }
    asm volatile("s_wait_asynccnt 0x0" ::: "memory");
  }
  __syncthreads();

  const int mrow = lane & 15;                // A: lane -> M
  const int khi  = lane >> 4;                // A/B: high half lanes hold K+2
  const int ncol = wave * 16 + (lane & 15);  // B: lane -> N

  v8f acc = {};
  #pragma unroll 4
  for (int k0 = 0; k0 < F_DIM; k0 += 4) {
    v2f a, b;
    // A 16x4: VGPR0 = K (k0 + 2*khi), VGPR1 = K+1
    a.x = xs[mrow * F_DIM + k0 + 2 * khi + 0];
    a.y = xs[mrow * F_DIM + k0 + 2 * khi + 1];
    // B 4x16: same K striping, lane indexes N
    b.x = w[(k0 + 2 * khi + 0) * F_DIM + ncol];
    b.y = w[(k0 + 2 * khi + 1) * F_DIM + ncol];
    acc = __builtin_amdgcn_wmma_f32_16x16x4_f32(
        /*neg_a=*/false, a, /*neg_b=*/false, b,
        /*c_mod=*/(short)0, acc, /*reuse_a=*/false, /*reuse_b=*/false);
  }

  // C/D layout: VGPR v -> row v (lanes 0-15) / row v+8 (lanes 16-31)
  const int rbase = block_m + khi * 8;
  #pragma unroll
  for (int v = 0; v < 8; ++v) {
    z[(rbase + v) * F_DIM + ncol] = acc[v];
  }
}

// el/er per (node, head): 32-elem dots against al/ar.
__global__ __launch_bounds__(256) void attn_scores(
    const float* __restrict__ z, const float* __restrict__ al,
    const float* __restrict__ ar, float* __restrict__ el,
    float* __restrict__ er) {
  int t = blockIdx.x * blockDim.x + threadIdx.x;
  if (t >= N_NODES * HEADS) return;
  int n = t >> 2, h = t & 3;
  const float* zp  = z  + n * F_DIM + h * D_HID;
  const float* alp = al + h * D_HID;
  const float* arp = ar + h * D_HID;
  float sl = 0.f, sr = 0.f;
  #pragma unroll
  for (int d = 0; d < D_HID; ++d) {
    float v = zp[d];
    sl += v * alp[d];
    sr += v * arp[d];
  }
  el[t] = sl;
  er[t] = sr;
}

// init mbuf (ord(-inf)), sbuf (0), agg (0)
__global__ __launch_bounds__(256) void init_bufs(
    unsigned* __restrict__ mbuf, float* __restrict__ sbuf,
    float* __restrict__ agg) {
  int t = blockIdx.x * blockDim.x + threadIdx.x;
  if (t < N_NODES * HEADS) {
    mbuf[t] = ORD_NEG_INF;
    sbuf[t] = 0.f;
  }
  if (t < N_NODES * F_DIM) agg[t] = 0.f;
}

// one thread per edge, all 4 heads: segment max via ordered-uint atomicMax
__global__ __launch_bounds__(256) void edge_max(
    const int* __restrict__ src, const int* __restrict__ dst,
    const float* __restrict__ el, const float* __restrict__ er,
    unsigned* __restrict__ mbuf, int nE) {
  int e = blockIdx.x * blockDim.x + threadIdx.x;
  if (e >= nE) return;
  int s = src[e], d = dst[e];
  float4 l = *(const float4*)(el + s * HEADS);
  float4 r = *(const float4*)(er + d * HEADS);
  unsigned* mp = mbuf + d * HEADS;
  atomicMax(mp + 0, ordf(lrelu(l.x + r.x)));
  atomicMax(mp + 1, ordf(lrelu(l.y + r.y)));
  atomicMax(mp + 2, ordf(lrelu(l.z + r.z)));
  atomicMax(mp + 3, ordf(lrelu(l.w + r.w)));
}

// one thread per edge, all 4 heads: segment sum of exp(score - max)
__global__ __launch_bounds__(256) void edge_expsum(
    const int* __restrict__ src, const int* __restrict__ dst,
    const float* __restrict__ el, const float* __restrict__ er,
    const unsigned* __restrict__ mbuf, float* __restrict__ sbuf, int nE) {
  int e = blockIdx.x * blockDim.x + threadIdx.x;
  if (e >= nE) return;
  int s = src[e], d = dst[e];
  float4 l = *(const float4*)(el + s * HEADS);
  float4 r = *(const float4*)(er + d * HEADS);
  const unsigned* mp = mbuf + d * HEADS;
  float* sp = sbuf + d * HEADS;
  atomicAdd(sp + 0, expf(lrelu(l.x + r.x) - unordf(mp[0])));
  atomicAdd(sp + 1, expf(lrelu(l.y + r.y) - unordf(mp[1])));
  atomicAdd(sp + 2, expf(lrelu(l.z + r.z) - unordf(mp[2])));
  atomicAdd(sp + 3, expf(lrelu(l.w + r.w) - unordf(mp[3])));
}

// one thread per (edge, 4-feature quad): float4 gather of z[src], 4 atomics
__global__ __launch_bounds__(256) void edge_aggregate(
    const int* __restrict__ src, const int* __restrict__ dst,
    const float* __restrict__ el, const float* __restrict__ er,
    const unsigned* __restrict__ mbuf, const float* __restrict__ sbuf,
    const float* __restrict__ z, float* __restrict__ agg, int nE) {
  int t = blockIdx.x * blockDim.x + threadIdx.x;
  if (t >= nE * (F_DIM / 4)) return;
  int e = t >> 5, q = t & 31;         // quad q covers features [4q, 4q+4)
  int h = q >> 3;                     // head = (4q)/32
  int s = src[e], d = dst[e];
  float xv = lrelu(el[s * HEADS + h] + er[d * HEADS + h]);
  float alpha = expf(xv - unordf(mbuf[d * HEADS + h])) / sbuf[d * HEADS + h];
  float4 zv = *(const float4*)(z + s * F_DIM + q * 4);
  float* ap = agg + d * F_DIM + q * 4;
  atomicAdd(ap + 0, zv.x * alpha);
  atomicAdd(ap + 1, zv.y * alpha);
  atomicAdd(ap + 2, zv.z * alpha);
  atomicAdd(ap + 3, zv.w * alpha);
}

// layer1 epilogue: h = relu(agg + b1)
__global__ __launch_bounds__(256) void bias_relu(
    const float* __restrict__ agg, const float* __restrict__ b,
    float* __restrict__ hout) {
  int t = blockIdx.x * blockDim.x + threadIdx.x;
  if (t >= N_NODES * F_DIM) return;
  float v = agg[t] + b[t & 127];
  hout[t] = v > 0.f ? v : 0.f;
}

// layer2 epilogue: out[n,d] = mean_h(agg[n, h*32+d] + b2[h*32+d])
__global__ __launch_bounds__(256) void bias_headmean(
    const float* __restrict__ agg, const float* __restrict__ b,
    float* __restrict__ out) {
  int t = blockIdx.x * blockDim.x + threadIdx.x;
  if (t >= N_NODES * D_HID) return;
  int n = t >> 5, d = t & 31;
  float acc = 0.f;
  #pragma unroll
  for (int h = 0; h < HEADS; ++h) {
    acc += agg[n * F_DIM + h * D_HID + d] + b[h * D_HID + d];
  }
  out[t] = acc * 0.25f;
}

extern "C" void kernel_launch(void* const* d_in, const int* in_sizes, int n_in,
                              void* d_out, int out_size, void* d_ws, size_t ws_size,
                              hipStream_t stream) {
  const float* feat = (const float*)d_in[0];
  const int*   src  = (const int*)d_in[1];
  const int*   dst  = (const int*)d_in[2];
  const float* W1   = (const float*)d_in[3];
  const float* al1  = (const float*)d_in[4];
  const float* ar1  = (const float*)d_in[5];
  const float* b1   = (const float*)d_in[6];
  const float* W2   = (const float*)d_in[7];
  const float* al2  = (const float*)d_in[8];
  const float* ar2  = (const float*)d_in[9];
  const float* b2   = (const float*)d_in[10];
  float* out = (float*)d_out;
  const int nE = in_sizes[1];

  // workspace layout (floats)
  float* bufA = (float*)d_ws;                     // N*128 : z1, then h, then agg2
  float* bufB = bufA + N_NODES * F_DIM;           // N*128 : agg1, then z2
  float* el   = bufB + N_NODES * F_DIM;           // N*4
  float* er   = el + N_NODES * HEADS;             // N*4
  unsigned* mbuf = (unsigned*)(er + N_NODES * HEADS);  // N*4
  float* sbuf = (float*)(mbuf + N_NODES * HEADS);      // N*4

  const int T = 256;
  const int gNH   = (N_NODES * HEADS + T - 1) / T;
  const int gNF   = (N_NODES * F_DIM + T - 1) / T;
  const int gE    = (nE + T - 1) / T;
  const int gEQ   = (nE * (F_DIM / 4) + T - 1) / T;
  const int gGemm = N_NODES / 16;  // 3125, exact

  // ---------------- Layer 1 ----------------
  gemm128_wmma_f32<<<gGemm, T, 0, stream>>>(feat, W1, bufA);          // z1 -> bufA
  attn_scores<<<gNH, T, 0, stream>>>(bufA, al1, ar1, el, er);
  init_bufs<<<gNF, T, 0, stream>>>(mbuf, sbuf, bufB);                 // agg1 -> bufB
  edge_max<<<gE, T, 0, stream>>>(src, dst, el, er, mbuf, nE);
  edge_expsum<<<gE, T, 0, stream>>>(src, dst, el, er, mbuf, sbuf, nE);
  edge_aggregate<<<gEQ, T, 0, stream>>>(src, dst, el, er, mbuf, sbuf, bufA, bufB, nE);
  bias_relu<<<gNF, T, 0, stream>>>(bufB, b1, bufA);                   // h -> bufA

  // ---------------- Layer 2 ----------------
  gemm128_wmma_f32<<<gGemm, T, 0, stream>>>(bufA, W2, bufB);          // z2 -> bufB
  attn_scores<<<gNH, T, 0, stream>>>(bufB, al2, ar2, el, er);
  init_bufs<<<gNF, T, 0, stream>>>(mbuf, sbuf, bufA);                 // agg2 -> bufA
  edge_max<<<gE, T, 0, stream>>>(src, dst, el, er, mbuf, nE);
  edge_expsum<<<gE, T, 0, stream>>>(src, dst, el, er, mbuf, sbuf, nE);
  edge_aggregate<<<gEQ, T, 0, stream>>>(src, dst, el, er, mbuf, sbuf, bufB, bufA, nE);
  bias_headmean<<<(N_NODES * D_HID + T - 1) / T, T, 0, stream>>>(bufA, b2, out);
}